// Actor_88862873354659
// MI455X (gfx1250) — compile-verified
//
#include <hip/hip_runtime.h>

namespace {
constexpr int Q = 1024, D = 2048, C = 128;
constexpr int TM = 32, TN = 64;          // tile: 32 a-rows x 64 b-rows
constexpr int RS = 132;                  // LDS row stride (floats): 16B aligned, bank-skewed
constexpr int THREADS = 256;             // 8 wave32
constexpr int OFF_FS  = Q * D;           // 2,097,152
constexpr int OFF_FB  = OFF_FS + Q * C;  // +131,072
constexpr int OFF_ACT = OFF_FB + D * C;  // +262,144
constexpr int NWG = (Q / TM) * (D / TN); // 1024 workgroups
}

typedef float f32x4 __attribute__((ext_vector_type(4)));
typedef int   i32x4 __attribute__((ext_vector_type(4)));

#if defined(__gfx1250__) && __has_builtin(__builtin_amdgcn_global_load_async_to_lds_b128) && __has_builtin(__builtin_amdgcn_s_wait_asynccnt)
#define USE_ASYNC 1
#else
#define USE_ASYNC 0
#endif

#if USE_ASYNC
typedef __attribute__((address_space(1))) i32x4* g_i32x4_p;
typedef __attribute__((address_space(3))) i32x4* l_i32x4_p;

__device__ __forceinline__ void async_b128(const float* g, float* l) {
  __builtin_amdgcn_global_load_async_to_lds_b128(
      (g_i32x4_p)(i32x4*)(void*)g,
      (l_i32x4_p)(i32x4*)(void*)l,
      /*offset=*/0, /*cpol=*/0);
}
#endif

__device__ __forceinline__ void argmax_upd(float v, int i, float& bv, int& bi) {
  // jnp.argmax tie-break: first (lowest flat index) wins on equality
  if (v > bv || (v == bv && i < bi)) { bv = v; bi = i; }
}

// ---------------------------------------------------------------------------
// Kernel 1: masked echo of fs / fb into the output buffer (vectorized,
// branch-light: one row-mask select per float4)
// ---------------------------------------------------------------------------
__global__ __launch_bounds__(256) void echo_kernel(
    const float* __restrict__ fea_s, const float* __restrict__ fea_b,
    const int* __restrict__ mask_s, const int* __restrict__ mask_b,
    float* __restrict__ out) {
  int i4 = blockIdx.x * blockDim.x + threadIdx.x;
  if (i4 < Q * C / 4) {
    f32x4 v = *(const f32x4*)(fea_s + i4 * 4);
    if (mask_s[i4 >> 5]) v = (f32x4){1.0f, 1.0f, 1.0f, 1.0f};
    *(f32x4*)(out + OFF_FS + i4 * 4) = v;
  }
  if (i4 < D * C / 4) {
    f32x4 v = *(const f32x4*)(fea_b + i4 * 4);
    if (mask_b[i4 >> 5]) v = (f32x4){-1.0f, -1.0f, -1.0f, -1.0f};
    *(f32x4*)(out + OFF_FB + i4 * 4) = v;
  }
}

// ---------------------------------------------------------------------------
// Kernel 2: tiled order-embedding score + per-workgroup argmax
// ---------------------------------------------------------------------------
__global__ __launch_bounds__(THREADS) void pred_kernel(
    const float* __restrict__ fea_s, const float* __restrict__ fea_b,
    const int* __restrict__ mask_s, const int* __restrict__ mask_b,
    float* __restrict__ out, float* __restrict__ wgmax, int* __restrict__ wgidx) {
  __shared__ __align__(16) float fsT[TM * RS];
  __shared__ __align__(16) float fbT[TN * RS];
  __shared__ int msS[TM];
  __shared__ int msB[TN];
  __shared__ float rdV[THREADS];
  __shared__ int   rdI[THREADS];

  const int tid   = threadIdx.x;
  const int aBase = blockIdx.y * TM;
  const int bBase = blockIdx.x * TN;

  // ---- stage tiles into LDS (async DMA on gfx1250) ----
#if USE_ASYNC
  #pragma unroll
  for (int it = 0; it < (TM * C / 4) / THREADS; ++it) {   // 4 iterations
    int i = tid + it * THREADS;
    int row = i >> 5, c4 = (i & 31) << 2;
    async_b128(fea_s + (aBase + row) * C + c4, fsT + row * RS + c4);
  }
  #pragma unroll
  for (int it = 0; it < (TN * C / 4) / THREADS; ++it) {   // 8 iterations
    int i = tid + it * THREADS;
    int row = i >> 5, c4 = (i & 31) << 2;
    async_b128(fea_b + (bBase + row) * C + c4, fbT + row * RS + c4);
  }
#else
  for (int i = tid; i < TM * C / 4; i += THREADS) {
    int row = i >> 5, c4 = (i & 31) << 2;
    *(f32x4*)(fsT + row * RS + c4) = *(const f32x4*)(fea_s + (aBase + row) * C + c4);
  }
  for (int i = tid; i < TN * C / 4; i += THREADS) {
    int row = i >> 5, c4 = (i & 31) << 2;
    *(f32x4*)(fbT + row * RS + c4) = *(const f32x4*)(fea_b + (bBase + row) * C + c4);
  }
#endif
  if (tid < TM) msS[tid] = mask_s[aBase + tid];
  if (tid < TN) msB[tid] = mask_b[bBase + tid];
#if USE_ASYNC
  __builtin_amdgcn_s_wait_asynccnt(0);
#endif
  __syncthreads();

  // ---- overwrite masked rows in LDS once (avoids per-FMA selects) ----
  for (int i = tid; i < TM * C; i += THREADS) {
    int row = i >> 7, c = i & (C - 1);
    if (msS[row]) fsT[row * RS + c] = 1.0f;
  }
  for (int i = tid; i < TN * C; i += THREADS) {
    int row = i >> 7, c = i & (C - 1);
    if (msB[row]) fbT[row * RS + c] = -1.0f;
  }
  __syncthreads();

  // ---- compute: 2 a-rows x 4 b-rows per thread ----
  const int tx  = tid & 15;   // b lane (b = bBase + tx + 16k) -> conflict-free banks
  const int ty  = tid >> 4;   // a group
  const int ar0 = ty * 2, ar1 = ar0 + 1;

  float acc[2][4] = {{0.f, 0.f, 0.f, 0.f}, {0.f, 0.f, 0.f, 0.f}};
  #pragma unroll 4
  for (int c4 = 0; c4 < C; c4 += 4) {
    f32x4 s0 = *(const f32x4*)(fsT + ar0 * RS + c4);
    f32x4 s1 = *(const f32x4*)(fsT + ar1 * RS + c4);
    #pragma unroll
    for (int k = 0; k < 4; ++k) {
      f32x4 bv = *(const f32x4*)(fbT + (tx + 16 * k) * RS + c4);
      float d;
      d = fmaxf(s0.x - bv.x, 0.f); acc[0][k] = fmaf(d, d, acc[0][k]);
      d = fmaxf(s0.y - bv.y, 0.f); acc[0][k] = fmaf(d, d, acc[0][k]);
      d = fmaxf(s0.z - bv.z, 0.f); acc[0][k] = fmaf(d, d, acc[0][k]);
      d = fmaxf(s0.w - bv.w, 0.f); acc[0][k] = fmaf(d, d, acc[0][k]);
      d = fmaxf(s1.x - bv.x, 0.f); acc[1][k] = fmaf(d, d, acc[1][k]);
      d = fmaxf(s1.y - bv.y, 0.f); acc[1][k] = fmaf(d, d, acc[1][k]);
      d = fmaxf(s1.z - bv.z, 0.f); acc[1][k] = fmaf(d, d, acc[1][k]);
      d = fmaxf(s1.w - bv.w, 0.f); acc[1][k] = fmaf(d, d, acc[1][k]);
    }
  }

  // ---- epilogue: pair-mask, store, thread-local argmax ----
  float bv = -__builtin_inff();
  int   bi = 0x7fffffff;
  #pragma unroll
  for (int ia = 0; ia < 2; ++ia) {
    int ar = ar0 + ia;
    int aG = aBase + ar;
    int mA = msS[ar];
    #pragma unroll
    for (int k = 0; k < 4; ++k) {
      int bT = tx + 16 * k;
      int bG = bBase + bT;
      float p = 1.0f - acc[ia][k];
      if (mA && msB[bT]) p = -1.0f;
      int flat = aG * D + bG;
      out[flat] = p;
      argmax_upd(p, flat, bv, bi);
    }
  }

  // ---- workgroup argmax tree ----
  rdV[tid] = bv; rdI[tid] = bi;
  __syncthreads();
  for (int s = THREADS / 2; s > 0; s >>= 1) {
    if (tid < s) argmax_upd(rdV[tid + s], rdI[tid + s], rdV[tid], rdI[tid]);
    __syncthreads();
  }
  if (tid == 0) {
    int wg = blockIdx.y * gridDim.x + blockIdx.x;
    wgmax[wg] = rdV[0];
    wgidx[wg] = rdI[0];
  }
}

// ---------------------------------------------------------------------------
// Kernel 3: final argmax over per-workgroup candidates -> action
// ---------------------------------------------------------------------------
__global__ __launch_bounds__(256) void final_kernel(
    const float* __restrict__ wgmax, const int* __restrict__ wgidx,
    float* __restrict__ out) {
  __shared__ float rv[256];
  __shared__ int   ri[256];
  int tid = threadIdx.x;
  float bv = -__builtin_inff();
  int   bi = 0x7fffffff;
  for (int i = tid; i < NWG; i += 256) argmax_upd(wgmax[i], wgidx[i], bv, bi);
  rv[tid] = bv; ri[tid] = bi;
  __syncthreads();
  for (int s = 128; s > 0; s >>= 1) {
    if (tid < s) argmax_upd(rv[tid + s], ri[tid + s], rv[tid], ri[tid]);
    __syncthreads();
  }
  if (tid == 0) {
    out[OFF_ACT]     = (float)(ri[0] / D);
    out[OFF_ACT + 1] = (float)(ri[0] % D);
  }
}

// ---------------------------------------------------------------------------
extern "C" void kernel_launch(void* const* d_in, const int* in_sizes, int n_in,
                              void* d_out, int out_size, void* d_ws, size_t ws_size,
                              hipStream_t stream) {
  (void)in_sizes; (void)n_in; (void)out_size; (void)ws_size;
  const float* fea_s = (const float*)d_in[0];
  const float* fea_b = (const float*)d_in[1];
  const int*   mask_s = (const int*)d_in[2];
  const int*   mask_b = (const int*)d_in[3];
  float* out = (float*)d_out;

  float* wgmax = (float*)d_ws;
  int*   wgidx = (int*)((char*)d_ws + NWG * sizeof(float));

  echo_kernel<<<(D * C / 4 + 255) / 256, 256, 0, stream>>>(fea_s, fea_b, mask_s, mask_b, out);

  dim3 grid(D / TN, Q / TM);  // 32 x 32 tiles
  pred_kernel<<<grid, THREADS, 0, stream>>>(fea_s, fea_b, mask_s, mask_b, out, wgmax, wgidx);

  final_kernel<<<1, 256, 0, stream>>>(wgmax, wgidx, out);
}